// ScaledDotProductAttention_56788057587918
// MI455X (gfx1250) — compile-verified
//
#include <hip/hip_runtime.h>

typedef __attribute__((ext_vector_type(2))) float v2f;
typedef __attribute__((ext_vector_type(8))) float v8f;

#define LSEQ 2048
#define DHEAD 64
#define NBH 16          // B*H
#define QTILE_WG 128    // q rows per workgroup (8 waves x 16)
#define KTILE 16        // k rows per main-loop iteration
#define NKT (LSEQ / KTILE)
#define SSTRIDE 18      // S-scratch row stride (floats)
#define KSTRIDE 68      // K-tile LDS row stride: b64 frag reads hit 64 distinct banks
#define VSTRIDE 72      // V-tile LDS row stride: half-waves on disjoint bank sets

// Truncating a flat __shared__ pointer yields the 32-bit LDS byte offset
// (LDS aperture base has zero low 32 bits).
#define LDS_OFF(p) ((unsigned)(unsigned long long)(p))

__global__ __launch_bounds__(256)
void attn_qkTv_f32_wmma(const float* __restrict__ q,
                        const float* __restrict__ k,
                        const float* __restrict__ v,
                        float* __restrict__ out) {
  __shared__ float sK[2][16 * KSTRIDE];   // staged K tiles (double buffered)
  __shared__ float sV[2][16 * VSTRIDE];   // staged V tiles (double buffered)
  __shared__ float sS[8][16 * SSTRIDE];   // per-wave S relayout scratch

  const int bid  = blockIdx.x;
  const int bh   = bid >> 4;          // 0..15  (b*H + h)
  const int qb   = bid & 15;          // 0..15  q block within sequence
  const int tid  = threadIdx.x;
  const int wave = tid >> 5;
  const int lane = tid & 31;
  const int lo   = lane & 15;         // M (A) / N (B,C,D) index
  const int hi   = lane >> 4;         // K-pair select (A,B); M+8 (C,D)

  const float* Q = q   + (size_t)bh * LSEQ * DHEAD;
  const float* K = k   + (size_t)bh * LSEQ * DHEAD;
  const float* V = v   + (size_t)bh * LSEQ * DHEAD;
  float*       O = out + (size_t)bh * LSEQ * DHEAD;

  const int q0 = qb * QTILE_WG + wave * 16;

  // ---- async staging: one b128 per thread per tile (256 thr = 4KB tile) ----
  const int srow = tid >> 4;          // 0..15 : tile row
  const int sc4  = (tid & 15) * 4;    // 0..60 : starting float within row
  auto issue_tile = [&](int kt, int buf) {
    const float* gk = K + (size_t)(kt * KTILE + srow) * DHEAD + sc4;
    const float* gv = V + (size_t)(kt * KTILE + srow) * DHEAD + sc4;
    unsigned lk = LDS_OFF(&sK[buf][srow * KSTRIDE + sc4]);
    unsigned lv = LDS_OFF(&sV[buf][srow * VSTRIDE + sc4]);
    asm volatile("global_load_async_to_lds_b128 %0, %1, off"
                 :: "v"(lk), "v"((unsigned long long)gk) : "memory");
    asm volatile("global_load_async_to_lds_b128 %0, %1, off"
                 :: "v"(lv), "v"((unsigned long long)gv) : "memory");
  };

  // ---- Q fragments resident in VGPRs for the whole kernel ----
  // A-layout (16x4 f32): lane(hi,lo): M=lo, vgpr0=A[M][4kk+2hi], vgpr1=+1
  v2f aq[16];
#pragma unroll
  for (int kk = 0; kk < 16; ++kk) {
    const float* p = Q + (size_t)(q0 + lo) * DHEAD + 4 * kk + 2 * hi;
    v2f a; a.x = p[0]; a.y = p[1];
    aq[kk] = a;
  }

  v8f zero8 = {0.f, 0.f, 0.f, 0.f, 0.f, 0.f, 0.f, 0.f};
  v8f oacc[4];
#pragma unroll
  for (int j = 0; j < 4; ++j) oacc[j] = zero8;

  float* sw = &sS[wave][0];

  issue_tile(0, 0);   // prologue: stage tile 0

  for (int kt = 0; kt < NKT; ++kt) {
    const int cur = kt & 1;
    // Prefetch next tile into the other buffer ((kt+1)&(NKT-1): uniform count,
    // last iteration harmlessly re-stages tile 0 -> wait value stays constant).
    issue_tile((kt + 1) & (NKT - 1), cur ^ 1);
    // Wait for *this* tile's 2 async loads (2 newer ones may stay in flight).
    asm volatile("s_wait_asynccnt 0x2" ::: "memory");
    __syncthreads();   // all waves' portions of tile kt are now in LDS

    // ---- GEMM 1: S(16x16) = Q_tile(16x64) x K_tile^T(64x16) ----
    // B-layout: lane(hi,lo): N=lo, vgpr0=B[2hi][N], vgpr1=B[2hi+1][N];
    // B = K^T so B[d][n] = Ktile[n][d] -> contiguous float2 in d from LDS.
    v8f s0 = zero8, s1 = zero8;   // two chains to shorten the RAW chain
#pragma unroll
    for (int kk = 0; kk < 8; ++kk) {
      const float* p0 = &sK[cur][lo * KSTRIDE + 4 * (2 * kk) + 2 * hi];
      const float* p1 = &sK[cur][lo * KSTRIDE + 4 * (2 * kk + 1) + 2 * hi];
      v2f b0; b0.x = p0[0]; b0.y = p0[1];
      v2f b1; b1.x = p1[0]; b1.y = p1[1];
      s0 = __builtin_amdgcn_wmma_f32_16x16x4_f32(
          false, aq[2 * kk], false, b0, (short)0, s0, false, false);
      s1 = __builtin_amdgcn_wmma_f32_16x16x4_f32(
          false, aq[2 * kk + 1], false, b1, (short)0, s1, false, false);
    }
    v8f s = s0 + s1;

    // ---- Relayout S via per-wave LDS scratch (C-layout -> A-layout) ----
#pragma unroll
    for (int r = 0; r < 8; ++r) {
      sw[(r + 8 * hi) * SSTRIDE + lo] = s[r];
    }
    // per-wave private region; DS ops are in-order within a wave

    // ---- GEMM 2: O(16x64) += S(16x16) x V_tile(16x64) ----
#pragma unroll
    for (int j = 0; j < 4; ++j) {
#pragma unroll
      for (int kk = 0; kk < 4; ++kk) {
        const float* pa = &sw[lo * SSTRIDE + 4 * kk + 2 * hi];
        v2f a2; a2.x = pa[0]; a2.y = pa[1];
        const int kd = 4 * kk + 2 * hi;
        const float* pv = &sV[cur][kd * VSTRIDE + j * 16 + lo];
        v2f b2; b2.x = pv[0]; b2.y = pv[VSTRIDE];
        oacc[j] = __builtin_amdgcn_wmma_f32_16x16x4_f32(
            false, a2, false, b2, (short)0, oacc[j], false, false);
      }
    }

    __syncthreads();   // everyone done reading buf 'cur' before it's re-staged
  }

  // ---- Store O: vgpr r of tile j -> out[q0 + r + 8*hi][j*16 + lo] ----
#pragma unroll
  for (int j = 0; j < 4; ++j) {
#pragma unroll
    for (int r = 0; r < 8; ++r) {
      O[(size_t)(q0 + r + 8 * hi) * DHEAD + j * 16 + lo] = oacc[j][r];
    }
  }
}

extern "C" void kernel_launch(void* const* d_in, const int* in_sizes, int n_in,
                              void* d_out, int out_size, void* d_ws, size_t ws_size,
                              hipStream_t stream) {
  const float* q = (const float*)d_in[0];
  const float* k = (const float*)d_in[1];
  const float* v = (const float*)d_in[2];
  float* out = (float*)d_out;

  dim3 grid(NBH * (LSEQ / QTILE_WG));  // 16 * 16 = 256 workgroups
  dim3 block(256);                     // 8 waves (wave32)
  attn_qkTv_f32_wmma<<<grid, block, 0, stream>>>(q, k, v, out);
}